// SSIM3D_1374389534921
// MI455X (gfx1250) — compile-verified
//
#include <hip/hip_runtime.h>
#include <math.h>

typedef __attribute__((ext_vector_type(2))) float v2f;
typedef __attribute__((ext_vector_type(8))) float v8f;

#define NB 2
#define DD 128
#define HH 256
#define WW 256
#define OD 26
#define OH 52
#define OW 52

static constexpr size_t FSZ = (size_t)NB * DD * OH * OW;   // 692224 floats per field
static constexpr int TOT_OUT = NB * OD * OH * OW;          // 140608 ssim voxels

// ---- Gaussian taps: g[t] = exp(-(t-5)^2/8)/sum, t in [0,10] --------------
__device__ __forceinline__ float gnorm_inv() {
  float s = 0.0f;
#pragma unroll
  for (int k = 0; k < 11; ++k) {
    float d = (float)(k - 5);
    s += expf(-d * d * 0.125f);
  }
  return 1.0f / s;
}

// banded conv matrix entry A[m][j] = g[j-5m] if 0<=j-5m<=10 else 0
__device__ __forceinline__ float a_entry(int m, int j, float invS) {
  int t = j - 5 * m;
  if (t < 0 || t > 10) return 0.0f;
  float d = (float)(t - 5);
  return expf(-d * d * 0.125f) * invS;
}

// ---- f32 WMMA wrapper: D = A(16x4) * B(4x16) + C ------------------------
__device__ __forceinline__ v8f wmma4(v2f a, v2f b, v8f c) {
  return __builtin_amdgcn_wmma_f32_16x16x4_f32(
      /*neg_a=*/false, a, /*neg_b=*/false, b,
      /*c_mod=*/(short)0, c, /*reuse_a=*/false, /*reuse_b=*/false);
}

// zero-padded global load along W
__device__ __forceinline__ float gload(const float* __restrict__ p, int iw) {
  int c = iw < 0 ? 0 : (iw > WW - 1 ? WW - 1 : iw);
  float v = p[c];
  return (iw == c) ? v : 0.0f;
}

// =========================================================================
// Stage A: fused W-conv + H-conv (both via V_WMMA_F32_16X16X4_F32)
// grid: (NB*DD)*4 blocks; block = 128 threads (4 waves)
// block bid -> slice = bid>>2 (n*128+d), t = bid&3 (ow block of 16)
// =========================================================================
__launch_bounds__(128)
__global__ void ssim_stageA(const float* __restrict__ img1,
                            const float* __restrict__ img2,
                            float* __restrict__ inter) {
  __shared__ float sV[5][112][17];   // [field][h-window row][ow-local], padded

  const int tid  = threadIdx.x;
  const int lane = tid & 31;
  const int wv   = tid >> 5;         // wave id 0..3
  const int bid  = blockIdx.x;
  const int t    = bid & 3;          // ow block
  const int slc  = bid >> 2;         // n*DD + d
  const size_t sbase = (size_t)slc * (HH * WW);

  const float invS = gnorm_inv();
  const int m16 = lane & 15;
  const int ko  = (lane >= 16) ? 2 : 0;   // K half-wave offset (A/B layout)
  const int mo  = (lane >= 16) ? 8 : 0;   // M half-wave offset (C/D layout)

  // Constant banded A matrix fragments, one v2f per K-step (shared by both convs)
  v2f amat[22];
#pragma unroll
  for (int s = 0; s < 22; ++s) {
    amat[s] = (v2f){a_entry(m16, 4 * s + ko, invS),
                    a_entry(m16, 4 * s + ko + 1, invS)};
  }

  const int wbase = 80 * t - 5;      // first input iw for this ow block

  for (int u = 0; u < 4; ++u) {      // oh blocks of 16
    const int hlo    = 80 * u - 5;                       // first h for this oh block
    const int h_lo_c = hlo < 0 ? 0 : hlo;
    int h_hi_c = hlo + 87;
    if (h_hi_c > HH - 1) h_hi_c = HH - 1;
    const int hb0   = h_lo_c >> 4;
    const int hb1   = h_hi_c >> 4;
    const int nhb   = hb1 - hb0 + 1;                     // 16h tiles needed (<=7)
    const int vbase = hb0 << 4;

    // ---------- Phase 1: W-conv tiles -> sV (waves split the h tiles) ----------
    for (int i = wv; i < nhb; i += 4) {
      const int h0  = vbase + (i << 4);
      const int row = h0 + m16;                          // lane's B column = image row h
      const float* p1 = img1 + sbase + (size_t)row * WW;
      const float* p2 = img2 + sbase + (size_t)row * WW;
      v8f a0 = {}; v8f a1 = {}; v8f a2 = {}; v8f a3 = {}; v8f a4 = {};
#pragma unroll
      for (int s = 0; s < 22; ++s) {
        const int j0  = 4 * s + ko;
        const int iw0 = wbase + j0;
        float x0 = gload(p1, iw0);
        float x1 = gload(p1, iw0 + 1);
        float y0 = gload(p2, iw0);
        float y1 = gload(p2, iw0 + 1);
        v2f bx  = (v2f){x0, x1};
        v2f by  = (v2f){y0, y1};
        v2f bxx = (v2f){x0 * x0, x1 * x1};
        v2f byy = (v2f){y0 * y0, y1 * y1};
        v2f bxy = (v2f){x0 * y0, x1 * y1};
        a0 = wmma4(amat[s], bx,  a0);
        a1 = wmma4(amat[s], by,  a1);
        a2 = wmma4(amat[s], bxx, a2);
        a3 = wmma4(amat[s], byy, a3);
        a4 = wmma4(amat[s], bxy, a4);
      }
      // D tile: reg r, this lane -> (M = r+mo = ow_local, N = m16 = h_local)
      const int vr = (h0 - vbase) + m16;
#pragma unroll
      for (int r = 0; r < 8; ++r) {
        sV[0][vr][r + mo] = a0[r];
        sV[1][vr][r + mo] = a1[r];
        sV[2][vr][r + mo] = a2[r];
        sV[3][vr][r + mo] = a3[r];
        sV[4][vr][r + mo] = a4[r];
      }
    }
    __syncthreads();

    // ---------- Phase 2: H-conv tile per field (waves split the 5 fields) ----------
    for (int f = wv; f < 5; f += 4) {
      v8f acc = {};
#pragma unroll
      for (int s = 0; s < 22; ++s) {
        const int j0  = 4 * s + ko;
        const int h0i = hlo + j0;
        const int h1i = h0i + 1;
        int c0 = h0i - vbase; c0 = c0 < 0 ? 0 : (c0 > 111 ? 111 : c0);
        int c1 = h1i - vbase; c1 = c1 < 0 ? 0 : (c1 > 111 ? 111 : c1);
        float b0 = sV[f][c0][m16];
        float b1 = sV[f][c1][m16];
        b0 = (h0i >= 0 && h0i <= h_hi_c) ? b0 : 0.0f;
        b1 = (h1i >= 0 && h1i <= h_hi_c) ? b1 : 0.0f;
        acc = wmma4(amat[s], (v2f){b0, b1}, acc);
      }
      const int ow = 16 * t + m16;
      if (ow < OW) {
#pragma unroll
        for (int r = 0; r < 8; ++r) {
          const int oh = 16 * u + r + mo;
          if (oh < OH) {
            inter[(size_t)f * FSZ + (((size_t)slc * OH + oh) * OW + ow)] = acc[r];
          }
        }
      }
    }
    __syncthreads();
  }
}

// =========================================================================
// Stage C: D-axis conv + SSIM map + per-block reduction
// =========================================================================
__launch_bounds__(256)
__global__ void ssim_stageC(const float* __restrict__ inter,
                            float* __restrict__ partials) {
  const int idx = blockIdx.x * 256 + threadIdx.x;
  float val = 0.0f;
  if (idx < TOT_OUT) {
    int ow = idx % OW;
    int r  = idx / OW;
    int oh = r % OH; r /= OH;
    int od = r % OD;
    int nb = r / OD;
    const float invS = gnorm_inv();
    float a0 = 0.f, a1 = 0.f, a2 = 0.f, a3 = 0.f, a4 = 0.f;
#pragma unroll
    for (int k = 0; k < 11; ++k) {
      int d = 5 * od - 5 + k;
      if (d >= 0 && d < DD) {
        float dk = (float)(k - 5);
        float gk = expf(-dk * dk * 0.125f) * invS;
        size_t off = (((size_t)(nb * DD + d)) * OH + oh) * OW + ow;
        a0 += gk * inter[off];
        a1 += gk * inter[FSZ + off];
        a2 += gk * inter[2 * FSZ + off];
        a3 += gk * inter[3 * FSZ + off];
        a4 += gk * inter[4 * FSZ + off];
      }
    }
    float s1  = a2 - a0 * a0;
    float s2  = a3 - a1 * a1;
    float s12 = a4 - a0 * a1;
    val = 2.0f * s12 / (s1 + s2);
  }
  __shared__ float red[256];
  red[threadIdx.x] = val;
  __syncthreads();
  for (int s = 128; s > 0; s >>= 1) {
    if (threadIdx.x < s) red[threadIdx.x] += red[threadIdx.x + s];
    __syncthreads();
  }
  if (threadIdx.x == 0) partials[blockIdx.x] = red[0];
}

// =========================================================================
// Stage D: deterministic final reduction, out = 1 - mean(ssim_map)
// =========================================================================
__launch_bounds__(256)
__global__ void ssim_stageD(const float* __restrict__ partials, int n,
                            float* __restrict__ out) {
  __shared__ float red[256];
  float v = 0.0f;
  for (int i = threadIdx.x; i < n; i += 256) v += partials[i];
  red[threadIdx.x] = v;
  __syncthreads();
  for (int s = 128; s > 0; s >>= 1) {
    if (threadIdx.x < s) red[threadIdx.x] += red[threadIdx.x + s];
    __syncthreads();
  }
  if (threadIdx.x == 0) out[0] = 1.0f - red[0] / (float)TOT_OUT;
}

// =========================================================================
extern "C" void kernel_launch(void* const* d_in, const int* in_sizes, int n_in,
                              void* d_out, int out_size, void* d_ws, size_t ws_size,
                              hipStream_t stream) {
  const float* img1 = (const float*)d_in[0];
  const float* img2 = (const float*)d_in[1];
  float* inter    = (float*)d_ws;                 // 5 * FSZ floats (~13.85 MB)
  float* partials = inter + 5 * FSZ;

  const int nblkC = (TOT_OUT + 255) / 256;        // 550

  ssim_stageA<<<dim3(NB * DD * 4), dim3(128), 0, stream>>>(img1, img2, inter);
  ssim_stageC<<<dim3(nblkC), dim3(256), 0, stream>>>(inter, partials);
  ssim_stageD<<<dim3(1), dim3(256), 0, stream>>>(partials, nblkC, (float*)d_out);
}